// DangoCutouts_9216999817696
// MI455X (gfx1250) — compile-verified
//
#include <hip/hip_runtime.h>
#include <hip/hip_bf16.h>

#ifndef __has_builtin
#define __has_builtin(x) 0
#endif

#define CUT        512
#define HH         4096
#define WW         4096
#define CH_STRIDE  (HH * WW)          // 16777216
#define PLANE      (CUT * CUT)        // 262144
#define TILES_PER_SLOT (PLANE / 16)   // 16384
#define GRAY_TILES (3 * TILES_PER_SLOT)

typedef __attribute__((ext_vector_type(2))) float v2f;
typedef __attribute__((ext_vector_type(8))) float v8f;

#if defined(__HIP_DEVICE_COMPILE__) && __has_builtin(__builtin_amdgcn_wmma_f32_16x16x4_f32)
#define HAVE_WMMA_F32X4 1
#else
#define HAVE_WMMA_F32X4 0
#endif

// ---------------------------------------------------------------------------
// Pass 1: bilinear crop-resize. grid = (512 rows, 13 sources), block = 512.
// src 0 = full-image resize -> slots 0 (rgb), 1 (rgb staged for gray),
//                              2 (x-flip rgb), 3 (flip rgb staged for gray)
// src s in 1..12 = inner crop s-1 -> slot 3+s (slot 4 staged rgb, grayed later)
// Each thread owns one output column: its x-tap coords live in registers,
// no LDS round trip; taps hit L2 (input ~ fits the 192MB L2).
// ---------------------------------------------------------------------------
__global__ __launch_bounds__(CUT) void dango_resample(
    const float* __restrict__ img,
    const int*   __restrict__ sizes,
    const int*   __restrict__ offys,
    const int*   __restrict__ offxs,
    float*       __restrict__ out)
{
    const int j   = blockIdx.x;   // output row
    const int src = blockIdx.y;   // 0 = full, 1..12 = inner crops
    const int i   = threadIdx.x;  // output col

    int size, offy, offx;
    if (src == 0) { size = 4096; offy = 0; offx = 0; }
    else          { size = sizes[src - 1]; offy = offys[src - 1]; offx = offxs[src - 1]; }

    const float s = (float)size;

    // x coordinate for this column (mirrors jnp op order exactly)
    float tx  = (i + 0.5f) * s / 512.0f - 0.5f;
    float xlo = (float)offx;
    float xhi = (float)offx + s - 1.0f;
    float x   = fminf(fmaxf(xlo + tx, xlo), xhi);
    int   x0  = (int)floorf(x);
    int   x1  = min(x0 + 1, offx + size - 1);
    float wx  = x - (float)x0;

    // y coordinate for this row
    float ty  = (j + 0.5f) * s / 512.0f - 0.5f;
    float ylo = (float)offy;
    float yhi = (float)offy + s - 1.0f;
    float y   = fminf(fmaxf(ylo + ty, ylo), yhi);
    int   y0  = (int)floorf(y);
    int   y1  = min(y0 + 1, offy + size - 1);
    float wy  = y - (float)y0;

    const float* r0 = img + (size_t)y0 * WW;
    const float* r1 = img + (size_t)y1 * WW;

    float vals[3];
#pragma unroll
    for (int c = 0; c < 3; ++c) {
        const float* p0 = r0 + (size_t)c * CH_STRIDE;
        const float* p1 = r1 + (size_t)c * CH_STRIDE;
        __builtin_prefetch(p1 + x0, 0, 0);  // global_prefetch_b8: warm second tap row
        float a = p0[x0], b = p0[x1];
        float d = p1[x0], e = p1[x1];
        float top = a * (1.0f - wx) + b * wx;
        float bot = d * (1.0f - wx) + e * wx;
        vals[c]   = top * (1.0f - wy) + bot * wy;
    }

    if (src == 0) {
        const int fi = (CUT - 1) - i;  // flipped column
#pragma unroll
        for (int c = 0; c < 3; ++c) {
            float v = vals[c];
            out[(size_t)(0 * 3 + c) * PLANE + j * CUT + i ] = v;  // full rgb
            out[(size_t)(1 * 3 + c) * PLANE + j * CUT + i ] = v;  // staged -> gray pass
            out[(size_t)(2 * 3 + c) * PLANE + j * CUT + fi] = v;  // flipped rgb
            out[(size_t)(3 * 3 + c) * PLANE + j * CUT + fi] = v;  // staged -> gray pass
        }
    } else {
        const int slot = 3 + src;  // inner crop k -> slot 4+k
#pragma unroll
        for (int c = 0; c < 3; ++c)
            out[(size_t)(slot * 3 + c) * PLANE + j * CUT + i] = vals[c];
    }
}

// ---------------------------------------------------------------------------
// Pass 2: in-place grayscale of slots {1, 3, 4} via V_WMMA_F32_16X16X4_F32.
//   D(16x16) = A(16x4) x B(4x16), every A row = (w_r, w_g, w_b, 0),
//   B[k][n] = channel k of pixel n  =>  D[m][n] = gray(pixel n) for all m.
// 32-bit A layout (ISA 7.12.2): V0 = K0 (lanes 0-15) | K2 (lanes 16-31),
//                               V1 = K1 | K3.  B mirrored (rows over lanes).
// So: lanes 0-15 load R into B.V0 and G into B.V1; lanes 16-31 load blue
// into B.V0 and 0 into B.V1.  No cross-lane permutes needed.
// One 16-pixel tile per wave32; EXEC is full at the WMMA (uniform guard).
// ---------------------------------------------------------------------------
__global__ __launch_bounds__(256) void dango_gray_wmma(float* __restrict__ out)
{
    const unsigned tid  = blockIdx.x * blockDim.x + threadIdx.x;
    const unsigned wave = tid >> 5;   // wave32
    const unsigned lane = tid & 31u;
    if (wave >= GRAY_TILES) return;   // wave-uniform

    const unsigned slotSel = wave / TILES_PER_SLOT;  // 0,1,2
    const unsigned slot    = (slotSel == 0) ? 1u : ((slotSel == 1) ? 3u : 4u);
    const unsigned p0      = (wave % TILES_PER_SLOT) * 16u;

    float* base = out + (size_t)slot * 3u * PLANE;

    const unsigned n   = lane & 15u;
    const bool     hiH = lane >= 16u;   // upper lane half
    const unsigned pix = p0 + n;

    // B matrix: V0 = R (lanes 0-15) | blue (lanes 16-31); V1 = G | 0
    float b0 = base[(size_t)(hiH ? 2u : 0u) * PLANE + pix];
    float gl = base[(size_t)1u * PLANE + pix];
    float b1 = hiH ? 0.0f : gl;

#if HAVE_WMMA_F32X4
    v2f A, B;
    A[0] = hiH ? 0.114f : 0.2989f;  // K0 | K2 weights
    A[1] = hiH ? 0.0f   : 0.587f;   // K1 | K3 weights
    B[0] = b0;
    B[1] = b1;
    v8f C = {};
    // 8 args: (neg_a, A, neg_b, B, c_mod, C, reuse_a, reuse_b)
    C = __builtin_amdgcn_wmma_f32_16x16x4_f32(false, A, false, B,
                                              (short)0, C, false, false);
    // lanes 0-15: C[v] = D[v][n]; rows 0..2 all hold gray(pixel n)
    if (!hiH) {
        base[(size_t)0u * PLANE + pix] = C[0];
        base[(size_t)1u * PLANE + pix] = C[1];
        base[(size_t)2u * PLANE + pix] = C[2];
    }
#else
    // VALU fallback (also what the host-side parse sees)
    if (!hiH) {
        float r  = b0;
        float bl = base[(size_t)2u * PLANE + pix];
        float g  = fmaf(0.2989f, r, fmaf(0.587f, gl, 0.114f * bl));
        base[(size_t)0u * PLANE + pix] = g;
        base[(size_t)1u * PLANE + pix] = g;
        base[(size_t)2u * PLANE + pix] = g;
    }
#endif
}

// ---------------------------------------------------------------------------
// Launch. Inputs (setup_inputs order): 0=input f32[1,3,4096,4096], 1=t (unused
// by reference), 2=sizes i32[12], 3=offy i32[12], 4=offx i32[12].
// Output: f32[16,3,512,512].
// ---------------------------------------------------------------------------
extern "C" void kernel_launch(void* const* d_in, const int* in_sizes, int n_in,
                              void* d_out, int out_size, void* d_ws, size_t ws_size,
                              hipStream_t stream) {
    (void)in_sizes; (void)n_in; (void)out_size; (void)d_ws; (void)ws_size;
    const float* img   = (const float*)d_in[0];
    const int*   sizes = (const int*)d_in[2];
    const int*   offy  = (const int*)d_in[3];
    const int*   offx  = (const int*)d_in[4];
    float*       out   = (float*)d_out;

    dim3 grid(CUT, 13);  // 512 rows x (1 full + 12 inner) sources
    dango_resample<<<grid, CUT, 0, stream>>>(img, sizes, offy, offx, out);

    const int grayBlocks = (GRAY_TILES * 32) / 256;  // 6144, exact
    dango_gray_wmma<<<grayBlocks, 256, 0, stream>>>(out);
}